// RLRNN_26456998543908
// MI455X (gfx1250) — compile-verified
//
#include <hip/hip_runtime.h>
#include <math.h>

typedef __attribute__((ext_vector_type(2))) float v2f;
typedef __attribute__((ext_vector_type(8))) float v8f;

#define S_   512
#define B_   8192
#define A_   2
#define H_   256
#define E_   8
#define P_   100
#define NSEQ (B_ * A_)   /* 16384 independent sequences */

// ---- fast tanh: gfx1250 V_TANH_F32 trans op (probe-confirmed in round 1) --
__device__ __forceinline__ float fast_tanh(float x) {
#if __has_builtin(__builtin_amdgcn_tanhf)
  return __builtin_amdgcn_tanhf(x);
#elif __has_builtin(__builtin_amdgcn_tanh_f32)
  return __builtin_amdgcn_tanh_f32(x);
#else
  float e = __expf(2.0f * x);
  return (e - 1.0f) * __builtin_amdgcn_rcpf(e + 1.0f);
#endif
}

// fold stage: keep my row-group's partial, send partner's group, add received.
__device__ __forceinline__ float foldx(float a, float b, bool hi, int mask) {
  float keep = hi ? b : a;
  float send = hi ? a : b;
  return keep + __shfl_xor(send, mask, 32);
}

// ---- kernel 1: baseW[p][j] = b1[j] + sum_e emb[p][e] * W1[2+e][j] ---------
__global__ __launch_bounds__(H_) void rlrnn_prep(
    const float* __restrict__ W1, const float* __restrict__ b1,
    const float* __restrict__ emb, float* __restrict__ baseW) {
  const int p = blockIdx.x;
  const int j = threadIdx.x;
  float acc = b1[j];
#pragma unroll
  for (int e = 0; e < E_; ++e)
    acc = fmaf(emb[p * E_ + e], W1[(2 + e) * H_ + j], acc);
  baseW[p * H_ + j] = acc;
}

// ---- kernel 2: the scan ----------------------------------------------------
// One wave = 16 sequences (8 b's x 2 actions). Row permutation: WMMA row M
// maps to sequence perm(M): rows 0-7 = (b0..b7, a=0), rows 8-15 = (b0..b7,
// a=1). Rows M and M+8 share a VGPR (lane halves) AND a participant id, so
// every C register is a single per-lane ds_load from the LDS baseW table.
// Per step, per 16-hidden tile:
//   D = A(16x4: [state, r, 0, 0]) x B(4x16: [w1row0; w1row1; 0; 0]) + C
// via V_WMMA_F32_16X16X4_F32, then h = relu(tanh(D)) (v_tanh_f32),
// upd = h.W2 via fold-reduce, new_state = sigmoid(state + (upd+b2)*mask).
__global__ __launch_bounds__(256) void rlrnn_scan(
    const float* __restrict__ cReward, const float* __restrict__ W1,
    const float* __restrict__ W2, const float* __restrict__ b2p,
    const int* __restrict__ cAction, const int* __restrict__ cPid,
    const float* __restrict__ baseW, float* __restrict__ out) {
  __shared__ float sBase[P_ * H_];   // 100 KB projected-embedding table

  const int tid = threadIdx.x;
  for (int i = tid; i < P_ * H_; i += 256) sBase[i] = baseW[i];
  __syncthreads();

  const int lane = tid & 31;
  const int wave = blockIdx.x * 8 + (tid >> 5);   // 0..1023
  const int seq0 = wave * 16;
  const int b0   = seq0 >> 1;                     // 8 b's per wave
  const bool low = lane < 16;
  const int l15  = lane & 15;
  // row M -> local sequence: M<8 -> 2M (a=0), M>=8 -> 2(M-8)+1 (a=1)
  const int pl   = (l15 & 8) ? ((l15 & 7) * 2 + 1) : (l15 * 2);
  const bool p0  = lane & 1, p1 = lane & 2, p2 = lane & 4, p3 = lane & 8;

  // B tiles (constant): VGPR0 = {K0 | K2=0}, VGPR1 = {K1 | K3=0}
  v2f  Bt[16];
  float w2l[16];
#pragma unroll
  for (int n = 0; n < 16; ++n) {
    const int col = n * 16 + l15;
    Bt[n][0] = low ? W1[0 * H_ + col] : 0.0f;   // K=0: state weights
    Bt[n][1] = low ? W1[1 * H_ + col] : 0.0f;   // K=1: reward weights
    w2l[n]   = W2[col];
  }
  const float b2 = b2p[0];

  float state = 0.0f;                 // lanes 0..15: state of seq0+perm(lane)

  // software-pipelined per-step operands (r, action, pid)
  float rC   = cReward[seq0 + pl];
  int   actC = cAction[seq0 + pl];
  int   pidC = cPid[b0 + (lane & 7)];

  for (int s = 0; s < S_; ++s) {
    const int base = s * NSEQ + seq0;

    // issue next step's loads now; they complete under the tile loop
    const int sn = (s + 1 < S_) ? s + 1 : s;
    const float rN   = cReward[sn * NSEQ + seq0 + pl];
    const int   actN = cAction[sn * NSEQ + seq0 + pl];
    const int   pidN = cPid[sn * B_ + b0 + (lane & 7)];

    // baseW row offset for each C register (same for both lane halves)
    int prow[8];
#pragma unroll
    for (int m = 0; m < 8; ++m)
      prow[m] = __shfl(pidC, m, 32) * H_;

    v2f a;
    a[0] = state;                // high lanes already 0 (K=2 slot)
    a[1] = low ? rC : 0.0f;      // K=1 / K=3 zero

    float acc[8];
#pragma unroll
    for (int m = 0; m < 8; ++m) acc[m] = 0.0f;

#pragma unroll
    for (int n = 0; n < 16; ++n) {
      const int col = n * 16 + l15;
      v8f c;
#pragma unroll
      for (int m = 0; m < 8; ++m) c[m] = sBase[prow[m] + col];
      v8f d = __builtin_amdgcn_wmma_f32_16x16x4_f32(
          false, a, false, Bt[n], (short)0, c, false, false);
#pragma unroll
      for (int m = 0; m < 8; ++m) {
        float h = fmaxf(fast_tanh(d[m]), 0.0f);
        acc[m] = fmaf(h, w2l[n], acc[m]);
      }
    }

    // fold-reduce: route row m's sum to lanes with (lane&7)==m while
    // reducing across the 16 lanes of each half.
    float f0 = foldx(acc[0], acc[1], p0, 1);
    float f1 = foldx(acc[2], acc[3], p0, 1);
    float f2 = foldx(acc[4], acc[5], p0, 1);
    float f3 = foldx(acc[6], acc[7], p0, 1);
    float g0 = foldx(f0, f1, p1, 2);
    float g1 = foldx(f2, f3, p1, 2);
    float h0 = foldx(g0, g1, p2, 4);
    h0 += __shfl_xor(h0, 8, 32);              // complete 16-lane sum
    float oth = __shfl_xor(h0, 16, 32);       // rows 8-15 live in high half
    const float upd = p3 ? oth : h0;          // lane l (<16) now holds row l

    const float msk = (actC == 1) ? 1.0f : 0.0f;
    const float z   = state + (upd + b2) * msk;
    const float ns  = __builtin_amdgcn_rcpf(1.0f + __expf(-z));
    state = low ? ns : 0.0f;
    if (low) out[base + pl] = ns;

    rC = rN; actC = actN; pidC = pidN;
  }
}

extern "C" void kernel_launch(void* const* d_in, const int* in_sizes, int n_in,
                              void* d_out, int out_size, void* d_ws, size_t ws_size,
                              hipStream_t stream) {
  const float* cReward = (const float*)d_in[0];
  const float* W1      = (const float*)d_in[1];
  const float* b1      = (const float*)d_in[2];
  const float* W2      = (const float*)d_in[3];
  const float* b2      = (const float*)d_in[4];
  const float* emb     = (const float*)d_in[5];
  const int*   cAction = (const int*)d_in[6];
  const int*   cPid    = (const int*)d_in[7];
  float* out   = (float*)d_out;
  float* baseW = (float*)d_ws;   // P_*H_ floats = 100 KB scratch

  rlrnn_prep<<<P_, H_, 0, stream>>>(W1, b1, emb, baseW);
  rlrnn_scan<<<NSEQ / 16 / 8, 256, 0, stream>>>(
      cReward, W1, W2, b2, cAction, cPid, baseW, out);
}